// QOC_75402445849120
// MI455X (gfx1250) — compile-verified
//
#include <hip/hip_runtime.h>

typedef float v2f __attribute__((ext_vector_type(2)));
typedef float v8f __attribute__((ext_vector_type(8)));

#define QD   64            // Hilbert dimension
#define QN   4096          // time steps
#define QL   8             // control channels
#define QDT  0.01f
#define MSZ  (QD*QD)       // 4096 floats per real 64x64 matrix
#define SLOT (2*MSZ)       // planar complex slot (re plane, im plane)
#define KTAY 12            // Taylor/Horner order (norm ~0.3 -> << fp32 eps)

// ---------------------------------------------------------------------------
// One 64x64 complex matmul, tiled 4x4 over 16 waves.
// Left operand stored TRANSPOSED in LDS (ATr/ATi: element A[m][k] at [k*64+m])
// so the 16x4 A-fragment load is lane-contiguous (conflict-free).
// Right operand row-major (Br/Bi: B[k][n] at [k*64+n]).
// Wave (tm,tn) produces the 16x16 tile; accumulators follow the CDNA5 C/D
// layout: reg r, lanes 0-15 -> M=r, lanes 16-31 -> M=r+8.
// Complex: cr = Ar*Br - Ai*Bi ; ci = Ar*Bi + Ai*Br  (4 WMMAs per K=4 slab).
// F32 WMMA has no A/B NEG modifier, so -Ai is formed in VALU.
// ---------------------------------------------------------------------------
__device__ __forceinline__ void cmatmul64(
    const float* __restrict__ ATr, const float* __restrict__ ATi,
    const float* __restrict__ Br,  const float* __restrict__ Bi,
    int tm, int tn, int lane, v8f& cr, v8f& ci)
{
  const int row  = tm*16 + (lane & 15);   // A row (M)
  const int col  = tn*16 + (lane & 15);   // B col (N)
  const int koff = (lane >> 4) << 1;      // lanes 0-15: K,K+1 ; lanes 16-31: K+2,K+3
  #pragma unroll
  for (int r = 0; r < 8; ++r) { cr[r] = 0.0f; ci[r] = 0.0f; }
  #pragma unroll
  for (int kb = 0; kb < 16; ++kb) {
    const int ka = kb*4 + koff;
    v2f ar, ai, br, bi;
    ar.x = ATr[(ka  )*QD + row];
    ar.y = ATr[(ka+1)*QD + row];
    ai.x = ATi[(ka  )*QD + row];
    ai.y = ATi[(ka+1)*QD + row];
    br.x = Br[(ka  )*QD + col];
    br.y = Br[(ka+1)*QD + col];
    bi.x = Bi[(ka  )*QD + col];
    bi.y = Bi[(ka+1)*QD + col];
    v2f nai = -ai;
    cr = __builtin_amdgcn_wmma_f32_16x16x4_f32(false, ar,  false, br, (short)0, cr, false, false);
    cr = __builtin_amdgcn_wmma_f32_16x16x4_f32(false, nai, false, bi, (short)0, cr, false, false);
    ci = __builtin_amdgcn_wmma_f32_16x16x4_f32(false, ar,  false, bi, (short)0, ci, false, false);
    ci = __builtin_amdgcn_wmma_f32_16x16x4_f32(false, ai,  false, br, (short)0, ci, false, false);
  }
}

// ---------------------------------------------------------------------------
// Pass 1: each block handles `chunk` consecutive time steps:
//   build A_i, expm via in-place Horner (S <- I + (A*S)/j), fold into the
//   running chunk product kept in per-wave accumulator registers.
// LDS: A^T (re/im) + S (re/im) = 64 KB exactly.
// ---------------------------------------------------------------------------
__global__ __launch_bounds__(512) void qoc_pass1(
    const float* __restrict__ Hd_r, const float* __restrict__ Hd_i,
    const float* __restrict__ Hc_r, const float* __restrict__ Hc_i,
    const float* __restrict__ act,  float* __restrict__ partials, int chunk)
{
  __shared__ float sATr[MSZ], sATi[MSZ], sBr[MSZ], sBi[MSZ];
  const int tid   = threadIdx.x;
  const int lane  = tid & 31;
  const int wave  = tid >> 5;
  const int tm    = wave >> 2, tn = wave & 3;
  const int mbase = tm*16 + ((lane >> 4) << 3);
  const int nc    = tn*16 + (lane & 15);
  const int g     = blockIdx.x;

  v8f pr, pi, cr, ci;
  #pragma unroll
  for (int r = 0; r < 8; ++r) { pr[r] = 0.0f; pi[r] = 0.0f; }

  for (int s = 0; s < chunk; ++s) {
    const int i = g*chunk + s;
    float coef[QL];
    #pragma unroll
    for (int l = 0; l < QL; ++l) coef[l] = act[l*QN + i];   // uniform -> s_loads

    // Build A^T = transpose(-i*DT*(Hd + sum_l a_l Hc_l)) and S = I.
    // A = Ar + i*Ai with Ar = DT*Im(M), Ai = -DT*Re(M).
    #pragma unroll
    for (int t = 0; t < 8; ++t) {
      const int e   = tid + t*512;
      const int row = e >> 6, col = e & 63;
      float mr = Hd_r[e], mi = Hd_i[e];
      #pragma unroll
      for (int l = 0; l < QL; ++l) {
        mr = fmaf(coef[l], Hc_r[l*MSZ + e], mr);
        mi = fmaf(coef[l], Hc_i[l*MSZ + e], mi);
      }
      sATr[col*QD + row] =  QDT * mi;
      sATi[col*QD + row] = -QDT * mr;
      sBr[e] = (row == col) ? 1.0f : 0.0f;
      sBi[e] = 0.0f;
    }
    __syncthreads();

    // Horner: S <- I + (A*S)/j , j = KTAY..1 ; in place (regs then barrier).
    for (int j = KTAY; j >= 1; --j) {
      cmatmul64(sATr, sATi, sBr, sBi, tm, tn, lane, cr, ci);
      __syncthreads();
      const float invj = 1.0f / (float)j;
      #pragma unroll
      for (int r = 0; r < 8; ++r) {
        const int M  = mbase + r;
        const float vr = fmaf(cr[r], invj, (M == nc) ? 1.0f : 0.0f);
        const float vi = ci[r] * invj;
        sBr[M*QD + nc] = vr;
        sBi[M*QD + nc] = vi;
        cr[r] = vr; ci[r] = vi;          // keep register copy of U tile
      }
      __syncthreads();
    }

    // Fold into running product (ordered: P <- P @ U).
    if (s == 0) {
      pr = cr; pi = ci;
    } else {
      // stash P (transposed) into the A slots, U is already in S slots
      #pragma unroll
      for (int r = 0; r < 8; ++r) {
        const int M = mbase + r;
        sATr[nc*QD + M] = pr[r];
        sATi[nc*QD + M] = pi[r];
      }
      __syncthreads();
      cmatmul64(sATr, sATi, sBr, sBi, tm, tn, lane, pr, pi);
      __syncthreads();
    }
  }

  // Write chunk partial product, planar complex.
  float* base = partials + (size_t)g * SLOT;
  #pragma unroll
  for (int r = 0; r < 8; ++r) {
    const int M = mbase + r;
    base[M*QD + nc]       = pr[r];
    base[MSZ + M*QD + nc] = pi[r];
  }
}

// ---------------------------------------------------------------------------
// Pass 2: ordered pairwise tree reduction: Out[k] = In[2k] @ In[2k+1].
// ---------------------------------------------------------------------------
__global__ __launch_bounds__(512) void qoc_reduce(
    const float* __restrict__ in, float* __restrict__ out)
{
  __shared__ float sATr[MSZ], sATi[MSZ], sBr[MSZ], sBi[MSZ];
  const int tid  = threadIdx.x;
  const int lane = tid & 31;
  const int wave = tid >> 5;
  const int tm   = wave >> 2, tn = wave & 3;
  const int k    = blockIdx.x;
  const float* X = in + (size_t)(2*k) * SLOT;   // left  (stored transposed)
  const float* Y = X + SLOT;                    // right (row-major)
  #pragma unroll
  for (int t = 0; t < 8; ++t) {
    const int e   = tid + t*512;
    const int row = e >> 6, col = e & 63;
    sATr[col*QD + row] = X[e];
    sATi[col*QD + row] = X[MSZ + e];
    sBr[e] = Y[e];
    sBi[e] = Y[MSZ + e];
  }
  __syncthreads();
  v8f cr, ci;
  cmatmul64(sATr, sATi, sBr, sBi, tm, tn, lane, cr, ci);
  float* ob = out + (size_t)k * SLOT;
  const int mbase = tm*16 + ((lane >> 4) << 3);
  const int nc    = tn*16 + (lane & 15);
  #pragma unroll
  for (int r = 0; r < 8; ++r) {
    const int M = mbase + r;
    ob[M*QD + nc]       = cr[r];
    ob[MSZ + M*QD + nc] = ci[r];
  }
}

// Planar -> interleaved complex64 output.
__global__ __launch_bounds__(512) void qoc_finalize(
    const float* __restrict__ in, float* __restrict__ out)
{
  const int tid = threadIdx.x;
  #pragma unroll
  for (int t = 0; t < 8; ++t) {
    const int e = tid + t*512;
    out[2*e]     = in[e];
    out[2*e + 1] = in[MSZ + e];
  }
}

extern "C" void kernel_launch(void* const* d_in, const int* in_sizes, int n_in,
                              void* d_out, int out_size, void* d_ws, size_t ws_size,
                              hipStream_t stream)
{
  const float* Hd_r = (const float*)d_in[0];
  const float* Hd_i = (const float*)d_in[1];
  const float* Hc_r = (const float*)d_in[2];
  const float* Hc_i = (const float*)d_in[3];
  const float* act  = (const float*)d_in[4];
  float* out = (float*)d_out;

  // Pick chunk so (m partials + m/2 ping-pong) fit in workspace.
  int chunk = 8;
  while (chunk < QN) {
    size_t m    = (size_t)QN / chunk;
    size_t need = (m + m/2) * SLOT * sizeof(float);
    if (need <= ws_size) break;
    chunk <<= 1;
  }
  int m = QN / chunk;
  float* bufA = (float*)d_ws;
  float* bufB = bufA + (size_t)m * SLOT;

  qoc_pass1<<<m, 512, 0, stream>>>(Hd_r, Hd_i, Hc_r, Hc_i, act, bufA, chunk);

  float* cin = bufA; float* cout = bufB;
  while (m > 1) {
    qoc_reduce<<<m/2, 512, 0, stream>>>(cin, cout);
    float* t = cin; cin = cout; cout = t;
    m >>= 1;
  }
  qoc_finalize<<<1, 512, 0, stream>>>(cin, out);

  (void)in_sizes; (void)n_in; (void)out_size;
}